// GCN_2843268350429
// MI455X (gfx1250) — compile-verified
//
#include <hip/hip_runtime.h>

#define NN 50000      // nodes
#define NE 800000     // edges
#define DH 128        // D_IN == D_HID
#define DO 64         // D_OUT
#define NG 128        // graphs

typedef float v2f __attribute__((ext_vector_type(2)));
typedef float v8f __attribute__((ext_vector_type(8)));

// ---------- degree / norm precompute ----------
__global__ void k_fill1(float* p, int n) {
    int i = blockIdx.x * blockDim.x + threadIdx.x;
    if (i < n) p[i] = 1.0f;
}

__global__ void k_deg_accum(const int* __restrict__ dst, float* __restrict__ deg, int E) {
    int e = blockIdx.x * blockDim.x + threadIdx.x;
    if (e < E) atomicAdd(&deg[dst[e]], 1.0f);
}

__global__ void k_deg_finish(float* __restrict__ deg_selfc, float* __restrict__ dinv, int n) {
    int i = blockIdx.x * blockDim.x + threadIdx.x;
    if (i < n) {
        float d = deg_selfc[i];
        dinv[i] = 1.0f / sqrtf(d);
        deg_selfc[i] = 1.0f / d;     // becomes self_coef in place
    }
}

__global__ void k_edge_w(const int* __restrict__ src, const int* __restrict__ dst,
                         const float* __restrict__ dinv, float* __restrict__ ew, int E) {
    int e = blockIdx.x * blockDim.x + threadIdx.x;
    if (e < E) ew[e] = dinv[src[e]] * dinv[dst[e]];
}

// ---------- WMMA fp32 GEMM: C[M,NC] = A[M,128] @ W[128,NC] (+bias)(+relu) ----------
// NC is compile-time so all LDS offsets fold into ds_load immediates.
// W is staged TRANSPOSED in LDS (sWt[c][k], padded stride 132) so each B
// fragment is a contiguous 8-byte pair, same as A -> ds_load_2addr_b64 both sides.
template <int NC>
__global__ void __launch_bounds__(256)
k_wmma_gemm(const float* __restrict__ A, const float* __restrict__ W,
            const float* __restrict__ bias, float* __restrict__ C,
            int M, int relu)
{
    __shared__ float sWt[NC * 132];   // W transposed: [c][k], stride 132
    __shared__ float sA[16 * 132];    // 16-row A strip, stride 132

    const int tid = threadIdx.x;
    const int r0  = blockIdx.x * 16;

    // stage W transposed: W is row-major [k][c]
    for (int i = tid; i < 128 * NC; i += 256) {
        int k = i / NC, c = i % NC;           // NC is power of 2 -> shift/mask
        sWt[c * 132 + k] = W[i];
    }
    // stage A strip
    for (int i = tid; i < 16 * 128; i += 256) {
        int r = i >> 7, c = i & 127;
        int row = r0 + r;
        sA[r * 132 + c] = (row < M) ? A[row * 128 + c] : 0.0f;
    }
    __syncthreads();

    const int wave = tid >> 5;
    const int lane = tid & 31;
    const int ln   = lane & 15;   // M (for A) / N (for B,D) index
    const int hi   = lane >> 4;   // K-half select (A/B), row-half select (D)
    const int c0   = wave * 16;

    if (c0 < NC) {                // wave-uniform: EXEC all-ones for WMMA
        const float* aP = &sA [ln        * 132 + 2 * hi];
        const float* bP = &sWt[(c0 + ln) * 132 + 2 * hi];
        v8f acc = {0.f, 0.f, 0.f, 0.f, 0.f, 0.f, 0.f, 0.f};
#pragma unroll
        for (int k = 0; k < 128; k += 4) {
            // A 16x4 f32 frag: lane=m, VGPRs = K {k+2hi, k+2hi+1}  (8B aligned)
            v2f a = *(const v2f*)(aP + k);
            // B 4x16 f32 frag: lane=n, same K striping (transposed LDS image)
            v2f b = *(const v2f*)(bP + k);
            acc = __builtin_amdgcn_wmma_f32_16x16x4_f32(
                false, a, false, b, (short)0, acc, false, false);
        }
        // D 16x16 f32: VGPR r holds row (r + 8*hi), col = ln
        const int col = c0 + ln;
        const float bv = bias ? bias[col] : 0.0f;
#pragma unroll
        for (int r = 0; r < 8; ++r) {
            int row = r0 + r + 8 * hi;
            if (row < M) {
                float v = acc[r] + bv;
                if (relu) v = fmaxf(v, 0.0f);
                C[row * NC + col] = v;
            }
        }
    }
}

// ---------- edge scatter-add: AGG[dst] += H2[src] * ew ----------
__global__ void k_scatter(const int* __restrict__ src, const int* __restrict__ dst,
                          const float* __restrict__ ew, const float* __restrict__ H2,
                          float* __restrict__ AGG, int E)
{
    long long idx = (long long)blockIdx.x * blockDim.x + threadIdx.x;
    if (idx >= (long long)E * DH) return;
    int e = (int)(idx >> 7);
    int d = (int)(idx & 127);
    atomicAdd(&AGG[(size_t)dst[e] * DH + d], H2[(size_t)src[e] * DH + d] * ew[e]);
}

// ---------- combine: H = relu(AGG + H2*self_coef + b) ----------
__global__ void k_combine(const float* __restrict__ AGG, const float* __restrict__ H2,
                          const float* __restrict__ selfc, const float* __restrict__ bias,
                          float* __restrict__ H, int n)
{
    int idx = blockIdx.x * blockDim.x + threadIdx.x;
    if (idx >= n * DH) return;
    int i = idx >> 7, d = idx & 127;
    float v = AGG[idx] + H2[idx] * selfc[i] + bias[d];
    H[idx] = fmaxf(v, 0.0f);
}

// ---------- global mean pool ----------
__global__ void k_pool_accum(const float* __restrict__ H, const int* __restrict__ batch,
                             float* __restrict__ sums, float* __restrict__ counts, int total)
{
    int idx = blockIdx.x * blockDim.x + threadIdx.x;
    if (idx >= total) return;
    int i = idx >> 7, d = idx & 127;
    int g = batch[i];
    atomicAdd(&sums[g * DH + d], H[idx]);
    if (d == 0) atomicAdd(&counts[g], 1.0f);
}

__global__ void k_pool_div(const float* __restrict__ sums, const float* __restrict__ counts,
                           float* __restrict__ pooled)
{
    int idx = blockIdx.x * blockDim.x + threadIdx.x;
    if (idx >= NG * DH) return;
    pooled[idx] = sums[idx] / fmaxf(counts[idx >> 7], 1.0f);
}

extern "C" void kernel_launch(void* const* d_in, const int* in_sizes, int n_in,
                              void* d_out, int out_size, void* d_ws, size_t ws_size,
                              hipStream_t stream)
{
    (void)in_sizes; (void)n_in; (void)out_size; (void)ws_size;

    const float* x     = (const float*)d_in[0];
    const int*   ei    = (const int*)d_in[1];
    const int*   src   = ei;
    const int*   dst   = ei + NE;
    const int*   batch = (const int*)d_in[2];
    // d_in[3] = num_graphs scalar (compile-time NG)
    const float* Wl[5] = {(const float*)d_in[4], (const float*)d_in[6], (const float*)d_in[8],
                          (const float*)d_in[10], (const float*)d_in[12]};
    const float* Bl[5] = {(const float*)d_in[5], (const float*)d_in[7], (const float*)d_in[9],
                          (const float*)d_in[11], (const float*)d_in[13]};
    const float* fc1_w = (const float*)d_in[14];
    const float* fc1_b = (const float*)d_in[15];
    const float* fc2_w = (const float*)d_in[16];
    const float* fc2_b = (const float*)d_in[17];

    // workspace carve-up (floats)
    float* H      = (float*)d_ws;            // NN*DH
    float* H2     = H      + (size_t)NN * DH;
    float* AGG    = H2     + (size_t)NN * DH;
    float* selfc  = AGG    + (size_t)NN * DH;   // deg -> self_coef in place
    float* dinv   = selfc  + NN;
    float* ew     = dinv   + NN;                // NE
    float* sums   = ew     + NE;                // NG*DH
    float* counts = sums   + (size_t)NG * DH;   // NG (contiguous with sums)
    float* pooled = counts + NG;                // NG*DH
    float* gact   = pooled + (size_t)NG * DH;   // NG*DH

    const int T = 256;

    // degree / norms (once)
    k_fill1<<<(NN + T - 1) / T, T, 0, stream>>>(selfc, NN);
    k_deg_accum<<<(NE + T - 1) / T, T, 0, stream>>>(dst, selfc, NE);
    k_deg_finish<<<(NN + T - 1) / T, T, 0, stream>>>(selfc, dinv, NN);
    k_edge_w<<<(NE + T - 1) / T, T, 0, stream>>>(src, dst, dinv, ew, NE);

    const long long escat = (long long)NE * DH;
    const int scat_blocks = (int)((escat + T - 1) / T);
    const int gemm_blocks = NN / 16;             // 50000 = 3125 * 16
    const int comb_blocks = (NN * DH + T - 1) / T;

    const float* cur = x;
    for (int l = 0; l < 5; ++l) {
        k_wmma_gemm<DH><<<gemm_blocks, T, 0, stream>>>(cur, Wl[l], nullptr, H2, NN, 0);
        hipMemsetAsync(AGG, 0, (size_t)NN * DH * sizeof(float), stream);
        k_scatter<<<scat_blocks, T, 0, stream>>>(src, dst, ew, H2, AGG, NE);
        k_combine<<<comb_blocks, T, 0, stream>>>(AGG, H2, selfc, Bl[l], H, NN);
        cur = H;
    }

    // mean pool
    hipMemsetAsync(sums, 0, ((size_t)NG * DH + NG) * sizeof(float), stream);
    k_pool_accum<<<(NN * DH + T - 1) / T, T, 0, stream>>>(H, batch, sums, counts, NN * DH);
    k_pool_div<<<(NG * DH + T - 1) / T, T, 0, stream>>>(sums, counts, pooled);

    // heads (K=128 both): fc1 + relu, fc2 -> d_out
    k_wmma_gemm<DH><<<NG / 16, T, 0, stream>>>(pooled, fc1_w, fc1_b, gact, NG, 1);
    k_wmma_gemm<DO><<<NG / 16, T, 0, stream>>>(gact, fc2_w, fc2_b, (float*)d_out, NG, 0);
}